// GCNEncoder_57655640981995
// MI455X (gfx1250) — compile-verified
//
#include <hip/hip_runtime.h>

#define NEG_SLOPE 0.01f

typedef __attribute__((ext_vector_type(2))) float v2f;
typedef __attribute__((ext_vector_type(8))) float v8f;

// ---------- fp32 hardware atomic add (GLOBAL_ATOMIC_ADD_F32, no CAS loop) ----
__device__ __forceinline__ void atomAddF(float* p, float v) {
    unsafeAtomicAdd(p, v);
}

// ---------- utility: grid-stride fill ---------------------------------------
__global__ void fill_f32(float* __restrict__ p, float v, long long n) {
    long long i = (long long)blockIdx.x * blockDim.x + threadIdx.x;
    long long stride = (long long)gridDim.x * blockDim.x;
    for (; i < n; i += stride) p[i] = v;
}

// ---------- degree count (in-degree incl. self-loop; deg pre-filled = 1) ----
__global__ void count_deg(const int* __restrict__ dst, float* __restrict__ deg, int E) {
    int e = blockIdx.x * blockDim.x + threadIdx.x;
    if (e < E) atomAddF(&deg[dst[e]], 1.0f);
}

__global__ void to_rsqrt(float* __restrict__ d, int n) {
    int i = blockIdx.x * blockDim.x + threadIdx.x;
    if (i < n) d[i] = rsqrtf(d[i]);   // deg >= 1 always (self-loop)
}

// ---------- WMMA f32 GEMM: Y[rows x FOUT] = X[rows x 256] @ W[256 x FOUT] ----
// One wave computes a 16x16 output tile; K loop in steps of 4 using
// V_WMMA_F32_16X16X4_F32. VGPR layouts per CDNA5 ISA 7.12.2:
//   A 16x4 f32 : lanes 0-15 -> M=lane, VGPR0/1 = K0/K1; lanes 16-31 -> K2/K3
//   B 4x16 f32 : N = lane%16, VGPR0/1 = (K0,K1) low half / (K2,K3) high half
//   C/D 16x16  : N = lane%16, VGPRr -> M=r (lanes 0-15) / M=r+8 (lanes 16-31)
template <int FOUT>
__global__ void gemm_wmma_f32(const float* __restrict__ X,
                              const float* __restrict__ W,
                              float* __restrict__ Y, int rows) {
    constexpr int TN = FOUT / 16;
    int wave = blockIdx.x * (blockDim.x >> 5) + (threadIdx.x >> 5);
    int tm = wave / TN;
    int tn = wave % TN;
    if (tm * 16 >= rows) return;          // wave-uniform: EXEC stays all-1s

    int lane = threadIdx.x & 31;
    int half = lane >> 4;                 // 0: K0,K1  1: K2,K3
    int lm   = lane & 15;

    const float* xr = X + (size_t)(tm * 16 + lm) * 256 + 2 * half;    // A row
    const float* wc = W + (size_t)(2 * half) * FOUT + tn * 16 + lm;   // B col

    v8f acc = {};
#pragma unroll 8
    for (int k = 0; k < 256; k += 4) {
        v2f a = *(const v2f*)(xr + k);    // 8B-aligned pair -> global_load_b64
        v2f b;
        b.x = wc[(size_t)k * FOUT];
        b.y = wc[(size_t)k * FOUT + FOUT];
        acc = __builtin_amdgcn_wmma_f32_16x16x4_f32(
            /*neg_a=*/false, a, /*neg_b=*/false, b,
            /*c_mod=*/(short)0, acc, /*reuse_a=*/false, /*reuse_b=*/false);
    }

    float* y = Y + (size_t)(tm * 16 + half * 8) * FOUT + tn * 16 + lm;
#pragma unroll
    for (int r = 0; r < 8; ++r) y[(size_t)r * FOUT] = acc[r];
}

// ---------- edge propagation, 256 features: wave per edge -------------------
__global__ void prop256(const float* __restrict__ h0,
                        const int* __restrict__ src, const int* __restrict__ dst,
                        const float* __restrict__ dis,
                        float* __restrict__ agg, int E) {
    int e = blockIdx.x * (blockDim.x >> 5) + (threadIdx.x >> 5);
    if (e >= E) return;
    int lane = threadIdx.x & 31;
    int s = src[e], d = dst[e];
    float nrm = dis[s] * dis[d];

    const float4* hp = (const float4*)(h0 + (size_t)s * 256) + lane;
    float4 v0 = hp[0];
    float4 v1 = hp[32];

    float* ap = agg + (size_t)d * 256 + lane * 4;
    atomAddF(ap + 0, v0.x * nrm); atomAddF(ap + 1, v0.y * nrm);
    atomAddF(ap + 2, v0.z * nrm); atomAddF(ap + 3, v0.w * nrm);
    float* ap2 = ap + 128;
    atomAddF(ap2 + 0, v1.x * nrm); atomAddF(ap2 + 1, v1.y * nrm);
    atomAddF(ap2 + 2, v1.z * nrm); atomAddF(ap2 + 3, v1.w * nrm);
}

// ---------- h = LeakyReLU(agg + h0*dis^2 (self-loop) + b), in place in agg --
__global__ void finalize_h(const float* __restrict__ h0, const float* __restrict__ b,
                           const float* __restrict__ dis, float* __restrict__ hagg,
                           long long n) {
    long long i = (long long)blockIdx.x * blockDim.x + threadIdx.x;
    if (i >= n) return;
    int node = (int)(i >> 8);
    int f = (int)(i & 255);
    float d2 = dis[node]; d2 *= d2;
    float v = hagg[i] + h0[i] * d2 + b[f];
    hagg[i] = (v >= 0.0f) ? v : NEG_SLOPE * v;
}

// ---------- fused edge propagation for mu & logvar (64 feats each) ----------
__global__ void prop64x2(const float* __restrict__ gmu, const float* __restrict__ glv,
                         const int* __restrict__ src, const int* __restrict__ dst,
                         const float* __restrict__ dis,
                         float* __restrict__ omu, float* __restrict__ olv, int E) {
    int e = blockIdx.x * (blockDim.x >> 5) + (threadIdx.x >> 5);
    if (e >= E) return;
    int lane = threadIdx.x & 31;
    int s = src[e], d = dst[e];
    float nrm = dis[s] * dis[d];

    float2 mv = ((const float2*)(gmu + (size_t)s * 64))[lane];
    float2 lv = ((const float2*)(glv + (size_t)s * 64))[lane];

    float* mo = omu + (size_t)d * 64 + lane * 2;
    float* lo = olv + (size_t)d * 64 + lane * 2;
    atomAddF(mo + 0, mv.x * nrm); atomAddF(mo + 1, mv.y * nrm);
    atomAddF(lo + 0, lv.x * nrm); atomAddF(lo + 1, lv.y * nrm);
}

// ---------- finalize outputs: bias + self-loop term; clamp logvar at 10 -----
__global__ void finalize_out(const float* __restrict__ gmu, const float* __restrict__ glv,
                             const float* __restrict__ bmu, const float* __restrict__ blv,
                             const float* __restrict__ dis,
                             float* __restrict__ omu, float* __restrict__ olv,
                             long long n) {
    long long i = (long long)blockIdx.x * blockDim.x + threadIdx.x;
    if (i >= n) return;
    int node = (int)(i >> 6);
    int f = (int)(i & 63);
    float d2 = dis[node]; d2 *= d2;
    omu[i] = omu[i] + gmu[i] * d2 + bmu[f];
    float lv = olv[i] + glv[i] * d2 + blv[f];
    olv[i] = fminf(lv, 10.0f);
}

// ---------------------------------------------------------------------------
extern "C" void kernel_launch(void* const* d_in, const int* in_sizes, int n_in,
                              void* d_out, int out_size, void* d_ws, size_t ws_size,
                              hipStream_t stream) {
    const float* x   = (const float*)d_in[0];
    const int*   ei  = (const int*)d_in[1];
    const float* Ws  = (const float*)d_in[2];
    const float* bs  = (const float*)d_in[3];
    const float* Wmu = (const float*)d_in[4];
    const float* bmu = (const float*)d_in[5];
    const float* Wlv = (const float*)d_in[6];
    const float* blv = (const float*)d_in[7];

    const int N = in_sizes[0] / 256;   // 100000
    const int E = in_sizes[1] / 2;     // 3200000
    const int* src = ei;
    const int* dst = ei + E;

    // workspace layout (floats): [dis | h0 (reused as gmu/glv) | hagg]
    float* ws = (float*)d_ws;
    const size_t Npad = ((size_t)N + 255) & ~(size_t)255;
    float* dis  = ws;
    float* h0   = ws + Npad;
    float* hagg = h0 + Npad * 256;
    float* gmu  = h0;                       // h0 dead after finalize_h
    float* glv  = h0 + (size_t)N * 64;

    float* omu = (float*)d_out;
    float* olv = omu + (size_t)N * 64;

    const int B   = 256;                    // 8 waves / block
    const int WPB = B / 32;

    // 1) deg := 1 (self-loop), zero accumulators (d_out is poisoned each run)
    fill_f32<<<1024, B, 0, stream>>>(dis, 1.0f, (long long)N);
    fill_f32<<<4096, B, 0, stream>>>(hagg, 0.0f, (long long)N * 256);
    fill_f32<<<4096, B, 0, stream>>>(omu, 0.0f, (long long)N * 128);

    // 2) in-degree count + rsqrt
    count_deg<<<(E + B - 1) / B, B, 0, stream>>>(dst, dis, E);
    to_rsqrt<<<(N + B - 1) / B, B, 0, stream>>>(dis, N);

    // 3) h0 = x @ W_shared   (N/16 * 256/16 = 100000 tiles, one wave each)
    {
        int tiles = (N / 16) * (256 / 16);
        gemm_wmma_f32<256><<<(tiles + WPB - 1) / WPB, B, 0, stream>>>(x, Ws, h0, N);
    }

    // 4) scatter normalized messages (256 feats)
    prop256<<<(E + WPB - 1) / WPB, B, 0, stream>>>(h0, src, dst, dis, hagg, E);

    // 5) h = LeakyReLU(agg + self-loop + bias), in place
    {
        long long n = (long long)N * 256;
        finalize_h<<<(unsigned)((n + B - 1) / B), B, 0, stream>>>(h0, bs, dis, hagg, n);
    }

    // 6) gmu = h @ W_mu ; glv = h @ W_lv
    {
        int tiles = (N / 16) * (64 / 16);
        gemm_wmma_f32<64><<<(tiles + WPB - 1) / WPB, B, 0, stream>>>(hagg, Wmu, gmu, N);
        gemm_wmma_f32<64><<<(tiles + WPB - 1) / WPB, B, 0, stream>>>(hagg, Wlv, glv, N);
    }

    // 7) fused scatter for mu & logvar
    prop64x2<<<(E + WPB - 1) / WPB, B, 0, stream>>>(gmu, glv, src, dst, dis, omu, olv, E);

    // 8) finalize outputs
    {
        long long n = (long long)N * 64;
        finalize_out<<<(unsigned)((n + B - 1) / B), B, 0, stream>>>(gmu, glv, bmu, blv, dis,
                                                                   omu, olv, n);
    }
}